// GraphPoolLayer_498216207037
// MI455X (gfx1250) — compile-verified
//
#include <hip/hip_runtime.h>
#include <stdint.h>

// ---------------------------------------------------------------------------
// GraphPool max-pooling over degree buckets (gfx1250 / MI455X)
//   out[r] = max(atom_features[self(r)], atom_features[adj[deg][p][0..deg-1]])
// HBM-bandwidth bound (~2 GB -> ~85us at 23.3 TB/s). No GEMM -> no WMMA; we
// exercise the CDNA5 data-movement engines instead:
//   * tensor_load_to_lds (TDM) for the block's contiguous 4KB of self rows
//   * global_load_async_to_lds_b128 for the block's adjacency index rows
//   * saddr-form global_load_b128 512B/wave coalesced neighbor-row gathers
//   * s_wait_asynccnt / s_wait_tensorcnt split counters, global_prefetch_b8
// ---------------------------------------------------------------------------

#define N_BUCKETS 11
#define BUCKET    50000
#define DIM       128
#define NATOMS    (N_BUCKETS * BUCKET)

#define ROWS_PER_BLOCK 8          // one wave32 per output row
#define THREADS        256        // 8 waves
static_assert(BUCKET % ROWS_PER_BLOCK == 0, "block must not straddle buckets");

#define ASG __attribute__((address_space(1)))
#define ASL __attribute__((address_space(3)))

typedef int          v4i   __attribute__((ext_vector_type(4)));
typedef unsigned int u32x4 __attribute__((ext_vector_type(4)));
typedef int          i32x8 __attribute__((ext_vector_type(8)));
typedef int          i32x4 __attribute__((ext_vector_type(4)));

#if defined(__HIP_DEVICE_COMPILE__) && __has_builtin(__builtin_amdgcn_global_load_async_to_lds_b128)
#define HAVE_ASYNC 1
#else
#define HAVE_ASYNC 0
#endif

#if defined(__HIP_DEVICE_COMPILE__) && __has_builtin(__builtin_amdgcn_tensor_load_to_lds)
#define HAVE_TDM 1
#else
#define HAVE_TDM 0
#endif

__device__ __forceinline__ float4 vmax4(float4 a, float4 b) {
  float4 r;
  r.x = __builtin_fmaxf(a.x, b.x);
  r.y = __builtin_fmaxf(a.y, b.y);
  r.z = __builtin_fmaxf(a.z, b.z);
  r.w = __builtin_fmaxf(a.w, b.w);
  return r;
}

// Wave-uniform gather of DEG neighbor rows; indices broadcast via readlane,
// all DEG b128 loads issued before the max chain for maximum MLP.
template <int DEG>
__device__ __forceinline__ float4 gatherMax(float4 m,
                                            const float* __restrict__ atoms,
                                            const int* idxRow,  // global or LDS
                                            int lane) {
  int myIdx = idxRow[lane < DEG ? lane : 0];
  int idx[DEG];
#pragma unroll
  for (int j = 0; j < DEG; ++j) idx[j] = __builtin_amdgcn_readlane(myIdx, j);

  float4 v[DEG];
#pragma unroll
  for (int j = 0; j < DEG; ++j)
    v[j] = *reinterpret_cast<const float4*>(atoms + (size_t)idx[j] * DIM + lane * 4);

#pragma unroll
  for (int j = 0; j < DEG; ++j) m = vmax4(m, v[j]);
  return m;
}

__global__ __launch_bounds__(THREADS) void graphpool_kernel(
    const float* __restrict__ atoms,     // [NATOMS, DIM]
    const int*   __restrict__ degSlice,  // [N_BUCKETS, 2]
    const int* __restrict__ a1,  const int* __restrict__ a2,
    const int* __restrict__ a3,  const int* __restrict__ a4,
    const int* __restrict__ a5,  const int* __restrict__ a6,
    const int* __restrict__ a7,  const int* __restrict__ a8,
    const int* __restrict__ a9,  const int* __restrict__ a10,
    float* __restrict__ out) {             // [NATOMS, DIM]
  const int tid  = threadIdx.x;
  const int lane = tid & 31;
  const int wave = tid >> 5;
  const int r0   = __builtin_amdgcn_readfirstlane(blockIdx.x * ROWS_PER_BLOCK);

  const int deg   = __builtin_amdgcn_readfirstlane(r0 / BUCKET);  // block-uniform bucket
  const int p0    = r0 - deg * BUCKET;   // first position inside bucket (multiple of 8)
  const int start = degSlice[2 * deg];   // bucket start row
  const int row   = r0 + wave;           // this wave's output row
  const int p     = p0 + wave;

  // Block-uniform adjacency base: rows p0..p0+7, i.e. 8*deg contiguous ints.
  const int* adjBase = nullptr;
  switch (deg) {
    case 1:  adjBase = a1  + (size_t)p0 * 1;  break;
    case 2:  adjBase = a2  + (size_t)p0 * 2;  break;
    case 3:  adjBase = a3  + (size_t)p0 * 3;  break;
    case 4:  adjBase = a4  + (size_t)p0 * 4;  break;
    case 5:  adjBase = a5  + (size_t)p0 * 5;  break;
    case 6:  adjBase = a6  + (size_t)p0 * 6;  break;
    case 7:  adjBase = a7  + (size_t)p0 * 7;  break;
    case 8:  adjBase = a8  + (size_t)p0 * 8;  break;
    case 9:  adjBase = a9  + (size_t)p0 * 9;  break;
    case 10: adjBase = a10 + (size_t)p0 * 10; break;
    default: break;
  }
  if (adjBase) __builtin_prefetch(adjBase, 0, 3);      // global_prefetch_b8

#if HAVE_ASYNC
  // Adjacency indices (8*deg ints = up to 320B, 32B-aligned) -> LDS, one
  // coalesced async b128 per lane, lanes 0..2*deg-1 of wave 0.
  __shared__ __align__(16) int idxbuf[ROWS_PER_BLOCK * 10];
  if (deg > 0 && tid < 2 * deg) {
    ASG v4i* g = (ASG v4i*)(uintptr_t)adjBase + tid;
    ASL v4i* l = (ASL v4i*)(uint32_t)(uintptr_t)idxbuf + tid;
    __builtin_amdgcn_global_load_async_to_lds_b128(g, l, 0, 0);
  }
#endif

#if HAVE_TDM
  // Self rows: 8 contiguous rows = 4KB tile; one TDM descriptor, wave 0 issues.
  // D# per CDNA5 ISA sec.8: 1024 x f32, 1 line, stride 1024.
  __shared__ __align__(16) float selfbuf[THREADS * 4];
  if (wave == 0) {
    uint64_t gaddr = (uint64_t)(uintptr_t)(atoms + (size_t)(start + p0) * DIM);
    uint32_t laddr = (uint32_t)(uintptr_t)selfbuf;
    u32x4 g0;
    g0.x = 1u;                                               // count=1 (valid, user mode)
    g0.y = laddr;                                            // lds_addr[31:0]
    g0.z = (uint32_t)gaddr;                                  // global_addr[31:0]
    g0.w = (uint32_t)((gaddr >> 32) & 0x1FFFFFFu) | (2u << 30); // addr[56:32] | type=2
    i32x8 g1;
    g1[0] = (int)(2u << 16);       // workgroup_mask=0, data_size=2 (4B)
    g1[1] = (int)(1024u << 16);    // tensor_dim0[15:0]=1024 @ bits[63:48]
    g1[2] = (int)(1u << 16);       // tensor_dim0[31:16]=0, tensor_dim1[15:0]=1
    g1[3] = (int)(1024u << 16);    // tensor_dim1[31:16]=0, tile_dim0=1024
    g1[4] = 1;                     // tile_dim1=1, tile_dim2=0
    g1[5] = 1024;                  // tensor_dim0_stride[31:0]=1024
    g1[6] = 0;                     // stride0[47:32]=0, tensor_dim1_stride[15:0]=0
    g1[7] = 0;
    i32x4 z4 = {0, 0, 0, 0};                   // groups 2/3 unused (<=2D tensor)
    i32x8 z8 = {0, 0, 0, 0, 0, 0, 0, 0};       // extra group (clang-23 6-arg form)
    __builtin_amdgcn_tensor_load_to_lds(g0, g1, z4, z4, z8, 0);
  }
#endif

  // Complete the copies, then share across waves.
#if HAVE_ASYNC
#if __has_builtin(__builtin_amdgcn_s_wait_asynccnt)
  __builtin_amdgcn_s_wait_asynccnt(0);
#else
  asm volatile("s_wait_asynccnt 0" ::: "memory");
#endif
#endif
#if HAVE_TDM
  if (wave == 0) {
#if __has_builtin(__builtin_amdgcn_s_wait_tensorcnt)
    __builtin_amdgcn_s_wait_tensorcnt(0);
#else
    asm volatile("s_wait_tensorcnt 0" ::: "memory");
#endif
  }
#endif
#if HAVE_ASYNC || HAVE_TDM
  __syncthreads();
#endif

  // ---- self row -----------------------------------------------------------
#if HAVE_TDM
  float4 m = reinterpret_cast<const float4*>(selfbuf)[tid];
#else
  float4 m = *reinterpret_cast<const float4*>(atoms + (size_t)(start + p) * DIM + lane * 4);
#endif

  // ---- neighbor gather + max ---------------------------------------------
#if HAVE_ASYNC
  const int* idxRow = idxbuf + wave * deg;              // LDS copy of this wave's indices
#else
  const int* idxRow = adjBase + (size_t)wave * deg;     // direct from global
#endif
  switch (deg) {
    case 1:  m = gatherMax<1>(m, atoms, idxRow, lane);  break;
    case 2:  m = gatherMax<2>(m, atoms, idxRow, lane);  break;
    case 3:  m = gatherMax<3>(m, atoms, idxRow, lane);  break;
    case 4:  m = gatherMax<4>(m, atoms, idxRow, lane);  break;
    case 5:  m = gatherMax<5>(m, atoms, idxRow, lane);  break;
    case 6:  m = gatherMax<6>(m, atoms, idxRow, lane);  break;
    case 7:  m = gatherMax<7>(m, atoms, idxRow, lane);  break;
    case 8:  m = gatherMax<8>(m, atoms, idxRow, lane);  break;
    case 9:  m = gatherMax<9>(m, atoms, idxRow, lane);  break;
    case 10: m = gatherMax<10>(m, atoms, idxRow, lane); break;
    default: break;  // deg 0: identity
  }

  *reinterpret_cast<float4*>(out + (size_t)row * DIM + lane * 4) = m;
}

// ---------------------------------------------------------------------------
extern "C" void kernel_launch(void* const* d_in, const int* in_sizes, int n_in,
                              void* d_out, int out_size, void* d_ws, size_t ws_size,
                              hipStream_t stream) {
  (void)in_sizes; (void)n_in; (void)out_size; (void)d_ws; (void)ws_size;
  const float* atoms    = (const float*)d_in[0];
  const int*   degSlice = (const int*)d_in[1];
  const int* a1  = (const int*)d_in[2];
  const int* a2  = (const int*)d_in[3];
  const int* a3  = (const int*)d_in[4];
  const int* a4  = (const int*)d_in[5];
  const int* a5  = (const int*)d_in[6];
  const int* a6  = (const int*)d_in[7];
  const int* a7  = (const int*)d_in[8];
  const int* a8  = (const int*)d_in[9];
  const int* a9  = (const int*)d_in[10];
  const int* a10 = (const int*)d_in[11];
  float* out = (float*)d_out;

  dim3 grid(NATOMS / ROWS_PER_BLOCK);   // 68750 blocks, exact
  graphpool_kernel<<<grid, THREADS, 0, stream>>>(
      atoms, degSlice, a1, a2, a3, a4, a5, a6, a7, a8, a9, a10, out);
}